// HTGNN_55920474193993
// MI455X (gfx1250) — compile-verified
//
#include <hip/hip_runtime.h>
#include <math.h>

#define N_NODES   100000
#define T_STEPS   10
#define F_DIM     64
#define E_EDGES   1600000
#define ED_DIM    8
#define H_DIM     128
#define B_BATCH   2048
#define C_CLS     2
#define BN_EPS    1e-5f

typedef float v2f __attribute__((ext_vector_type(2)));
typedef float v8f __attribute__((ext_vector_type(8)));

static __device__ __forceinline__ v8f wmma4(v2f a, v2f b, v8f c) {
    return __builtin_amdgcn_wmma_f32_16x16x4_f32(false, a, false, b, (short)0, c, false, false);
}

// ---------------------------------------------------------------- degree ----
__global__ void k_degree(const int* __restrict__ dst, float* __restrict__ deg) {
    int e = blockIdx.x * blockDim.x + threadIdx.x;
    if (e < E_EDGES) atomicAdd(&deg[dst[e]], 1.0f);
}

// ---------------- pack [w_msg_x | w_self] transposed: w_pack[col][k] -------
__global__ void k_pack_w(const float* __restrict__ w_msg, const float* __restrict__ w_self,
                         float* __restrict__ w_pack) {
    int idx = blockIdx.x * blockDim.x + threadIdx.x;   // 256*64
    if (idx >= 256 * F_DIM) return;
    int col = idx >> 6;        // 0..255
    int k   = idx & 63;        // 0..63
    float v = (col < H_DIM) ? w_msg[k * H_DIM + col] : w_self[k * H_DIM + (col - H_DIM)];
    w_pack[col * F_DIM + k] = v;
}

// ------------------------------------------ time-invariant edge messages ----
__global__ void k_edge_msg(const int* __restrict__ dst,
                           const float* __restrict__ edge_attr,
                           const float* __restrict__ w_msg,
                           float* __restrict__ agg_e) {
    long long idx = (long long)blockIdx.x * blockDim.x + threadIdx.x;
    if (idx >= (long long)E_EDGES * H_DIM) return;
    int e = (int)(idx >> 7);
    int h = (int)(idx & 127);
    const float* ea = edge_attr + (long long)e * ED_DIM;
    float s = 0.f;
#pragma unroll
    for (int k = 0; k < ED_DIM; ++k)
        s += ea[k] * w_msg[(F_DIM + k) * H_DIM + h];
    atomicAdd(&agg_e[(long long)dst[e] * H_DIM + h], s);
}

// ------------- OUT(N x 256) = x_t(N x 64) @ w_pack^T, 16x64 strip/wave -----
__global__ void k_gemm_x(const float* __restrict__ x,
                         const float* __restrict__ w_pack,
                         float* __restrict__ outbuf, int t) {
    int wave = (blockIdx.x * blockDim.x + threadIdx.x) >> 5;
    int lane = threadIdx.x & 31;
    int grp    = wave & 3;             // 4 column groups of 64
    int tile_m = wave >> 2;
    if (tile_m >= N_NODES / 16) return;
    int lane_lo = lane & 15;
    int hi = lane >> 4;
    int m0 = tile_m * 16;
    int col0 = grp * 64 + lane_lo;
    const float* arow = x + (size_t)(m0 + lane_lo) * (T_STEPS * F_DIM) + t * F_DIM;
    const float* bp0 = w_pack + (size_t)(col0)      * F_DIM;
    const float* bp1 = w_pack + (size_t)(col0 + 16) * F_DIM;
    const float* bp2 = w_pack + (size_t)(col0 + 32) * F_DIM;
    const float* bp3 = w_pack + (size_t)(col0 + 48) * F_DIM;
    v8f acc0 = {}, acc1 = {}, acc2 = {}, acc3 = {};
#pragma unroll
    for (int kk = 0; kk < F_DIM / 4; ++kk) {
        int k0 = kk * 4 + hi * 2;                  // always even -> 8B aligned
        v2f a  = *(const v2f*)(arow + k0);
        v2f b0 = *(const v2f*)(bp0 + k0);
        v2f b1 = *(const v2f*)(bp1 + k0);
        v2f b2 = *(const v2f*)(bp2 + k0);
        v2f b3 = *(const v2f*)(bp3 + k0);
        acc0 = wmma4(a, b0, acc0);
        acc1 = wmma4(a, b1, acc1);
        acc2 = wmma4(a, b2, acc2);
        acc3 = wmma4(a, b3, acc3);
    }
#pragma unroll
    for (int r = 0; r < 8; ++r) {
        size_t ro = (size_t)(m0 + r + 8 * hi) * 256;
        outbuf[ro + col0]      = acc0[r];
        outbuf[ro + col0 + 16] = acc1[r];
        outbuf[ro + col0 + 32] = acc2[r];
        outbuf[ro + col0 + 48] = acc3[r];
    }
}

// ------------------------------------------ scatter y_t[src] into agg ------
__global__ void k_scatter(const int* __restrict__ src, const int* __restrict__ dst,
                          const float* __restrict__ outbuf, float* __restrict__ agg) {
    long long idx = (long long)blockIdx.x * blockDim.x + threadIdx.x;
    if (idx >= (long long)E_EDGES * H_DIM) return;
    int e = (int)(idx >> 7);
    int h = (int)(idx & 127);
    float v = outbuf[(long long)src[e] * 256 + h];
    atomicAdd(&agg[(long long)dst[e] * H_DIM + h], v);
}

// ------- out = self + b_self + agg/cnt + (deg>0 ? b_msg : 0); BN stats -----
__global__ void k_combine(float* __restrict__ outbuf, const float* __restrict__ agg,
                          const float* __restrict__ deg,
                          const float* __restrict__ b_msg, const float* __restrict__ b_self,
                          float* __restrict__ bn_sum, float* __restrict__ bn_sumsq) {
    int h = threadIdx.x;               // 128 threads, one channel each
    int n0 = blockIdx.x * 64;
    float bs = b_self[h], bm = b_msg[h];
    float s = 0.f, ss = 0.f;
    for (int r = 0; r < 64; ++r) {
        int n = n0 + r;
        if (n >= N_NODES) break;
        float d = deg[n];
        float inv = 1.0f / fmaxf(d, 1.0f);
        float v = outbuf[(size_t)n * 256 + H_DIM + h] + bs
                + agg[(size_t)n * H_DIM + h] * inv
                + (d > 0.5f ? bm : 0.0f);
        outbuf[(size_t)n * 256 + h] = v;   // reuse y-half (scatter already done)
        s += v; ss += v * v;
    }
    atomicAdd(&bn_sum[h], s);
    atomicAdd(&bn_sumsq[h], ss);
}

__global__ void k_bnfin(const float* __restrict__ bn_sum, const float* __restrict__ bn_sumsq,
                        const float* __restrict__ gamma, const float* __restrict__ beta,
                        float* __restrict__ scale, float* __restrict__ shift) {
    int h = threadIdx.x;
    float mean = bn_sum[h] * (1.0f / (float)N_NODES);
    float var  = bn_sumsq[h] * (1.0f / (float)N_NODES) - mean * mean;
    float inv  = rsqrtf(var + BN_EPS);
    float sc   = inv * gamma[h];
    scale[h] = sc;
    shift[h] = beta[h] - mean * sc;
}

__global__ void k_focal(const float* __restrict__ outbuf, const int* __restrict__ ptr,
                        const float* __restrict__ scale, const float* __restrict__ shift,
                        float* __restrict__ hseq_t) {
    int idx = blockIdx.x * blockDim.x + threadIdx.x;
    if (idx >= B_BATCH * H_DIM) return;
    int b = idx >> 7, h = idx & 127;
    int n = ptr[b];
    float v = outbuf[(size_t)n * 256 + h] * scale[h] + shift[h];
    hseq_t[idx] = fmaxf(v, 0.0f);
}

// ------------------- G(T*B x 512) = h_seq(T*B x 128) @ w_ih^T --------------
__global__ void k_gemm_ih(const float* __restrict__ hseq, const float* __restrict__ w_ih,
                          float* __restrict__ G) {
    int wave = (blockIdx.x * blockDim.x + threadIdx.x) >> 5;
    int lane = threadIdx.x & 31;
    int grp    = wave & 7;             // 8 column groups of 64 (512 cols)
    int tile_m = wave >> 3;
    if (tile_m >= (T_STEPS * B_BATCH) / 16) return;
    int lane_lo = lane & 15, hi = lane >> 4;
    int m0 = tile_m * 16;
    int col0 = grp * 64 + lane_lo;
    const float* arow = hseq + (size_t)(m0 + lane_lo) * H_DIM;
    const float* bp0 = w_ih + (size_t)(col0)      * H_DIM;   // B[k][col] = w_ih[col][k]
    const float* bp1 = w_ih + (size_t)(col0 + 16) * H_DIM;
    const float* bp2 = w_ih + (size_t)(col0 + 32) * H_DIM;
    const float* bp3 = w_ih + (size_t)(col0 + 48) * H_DIM;
    v8f acc0 = {}, acc1 = {}, acc2 = {}, acc3 = {};
#pragma unroll 8
    for (int kk = 0; kk < H_DIM / 4; ++kk) {
        int k0 = kk * 4 + hi * 2;
        v2f a  = *(const v2f*)(arow + k0);
        v2f b0 = *(const v2f*)(bp0 + k0);
        v2f b1 = *(const v2f*)(bp1 + k0);
        v2f b2 = *(const v2f*)(bp2 + k0);
        v2f b3 = *(const v2f*)(bp3 + k0);
        acc0 = wmma4(a, b0, acc0);
        acc1 = wmma4(a, b1, acc1);
        acc2 = wmma4(a, b2, acc2);
        acc3 = wmma4(a, b3, acc3);
    }
#pragma unroll
    for (int r = 0; r < 8; ++r) {
        size_t ro = (size_t)(m0 + r + 8 * hi) * 512;
        G[ro + col0]      = acc0[r];
        G[ro + col0 + 16] = acc1[r];
        G[ro + col0 + 32] = acc2[r];
        G[ro + col0 + 48] = acc3[r];
    }
}

// -------- gates(B x 512) = h @ w_hh^T + G_t + b_ih + b_hh ------------------
__global__ void k_gemm_gates(const float* __restrict__ h, const float* __restrict__ w_hh,
                             const float* __restrict__ G_t,
                             const float* __restrict__ b_ih, const float* __restrict__ b_hh,
                             float* __restrict__ gates) {
    int wave = (blockIdx.x * blockDim.x + threadIdx.x) >> 5;
    int lane = threadIdx.x & 31;
    int grp    = wave & 7;
    int tile_m = wave >> 3;
    if (tile_m >= B_BATCH / 16) return;
    int lane_lo = lane & 15, hi = lane >> 4;
    int m0 = tile_m * 16;
    int col0 = grp * 64 + lane_lo;
    const float* arow = h + (size_t)(m0 + lane_lo) * H_DIM;
    const float* bp0 = w_hh + (size_t)(col0)      * H_DIM;
    const float* bp1 = w_hh + (size_t)(col0 + 16) * H_DIM;
    const float* bp2 = w_hh + (size_t)(col0 + 32) * H_DIM;
    const float* bp3 = w_hh + (size_t)(col0 + 48) * H_DIM;
    v8f acc0 = {}, acc1 = {}, acc2 = {}, acc3 = {};
#pragma unroll 8
    for (int kk = 0; kk < H_DIM / 4; ++kk) {
        int k0 = kk * 4 + hi * 2;
        v2f a  = *(const v2f*)(arow + k0);
        v2f b0 = *(const v2f*)(bp0 + k0);
        v2f b1 = *(const v2f*)(bp1 + k0);
        v2f b2 = *(const v2f*)(bp2 + k0);
        v2f b3 = *(const v2f*)(bp3 + k0);
        acc0 = wmma4(a, b0, acc0);
        acc1 = wmma4(a, b1, acc1);
        acc2 = wmma4(a, b2, acc2);
        acc3 = wmma4(a, b3, acc3);
    }
    float bias0 = b_ih[col0]      + b_hh[col0];
    float bias1 = b_ih[col0 + 16] + b_hh[col0 + 16];
    float bias2 = b_ih[col0 + 32] + b_hh[col0 + 32];
    float bias3 = b_ih[col0 + 48] + b_hh[col0 + 48];
#pragma unroll
    for (int r = 0; r < 8; ++r) {
        size_t ro = (size_t)(m0 + r + 8 * hi) * 512;
        gates[ro + col0]      = acc0[r] + G_t[ro + col0]      + bias0;
        gates[ro + col0 + 16] = acc1[r] + G_t[ro + col0 + 16] + bias1;
        gates[ro + col0 + 32] = acc2[r] + G_t[ro + col0 + 32] + bias2;
        gates[ro + col0 + 48] = acc3[r] + G_t[ro + col0 + 48] + bias3;
    }
}

__device__ __forceinline__ float sigf(float x) { return 1.0f / (1.0f + expf(-x)); }

__global__ void k_lstm(const float* __restrict__ gates,
                       float* __restrict__ hbuf, float* __restrict__ cbuf) {
    int idx = blockIdx.x * blockDim.x + threadIdx.x;
    if (idx >= B_BATCH * H_DIM) return;
    int b = idx >> 7, h = idx & 127;
    const float* g = gates + (size_t)b * 512;
    float i  = sigf(g[h]);
    float f  = sigf(g[H_DIM + h]);
    float gg = tanhf(g[2 * H_DIM + h]);
    float o  = sigf(g[3 * H_DIM + h]);
    float c  = f * cbuf[idx] + i * gg;
    cbuf[idx] = c;
    hbuf[idx] = o * tanhf(c);
}

__global__ void k_cls(const float* __restrict__ h, const float* __restrict__ w_cls,
                      const float* __restrict__ b_cls, float* __restrict__ out) {
    int idx = blockIdx.x * blockDim.x + threadIdx.x;
    if (idx >= B_BATCH * C_CLS) return;
    int b = idx >> 1, c = idx & 1;
    float s = b_cls[c];
#pragma unroll 16
    for (int k = 0; k < H_DIM; ++k)
        s += h[b * H_DIM + k] * w_cls[k * C_CLS + c];
    out[idx] = s;
}

// ---------------------------------------------------------------------------
extern "C" void kernel_launch(void* const* d_in, const int* in_sizes, int n_in,
                              void* d_out, int out_size, void* d_ws, size_t ws_size,
                              hipStream_t stream) {
    (void)in_sizes; (void)n_in; (void)out_size; (void)ws_size;

    const float* x        = (const float*)d_in[0];
    const int*   eidx     = (const int*)  d_in[1];
    const float* eattr    = (const float*)d_in[2];
    const int*   ptr      = (const int*)  d_in[3];
    const float* w_msg    = (const float*)d_in[4];
    const float* b_msg    = (const float*)d_in[5];
    const float* w_self   = (const float*)d_in[6];
    const float* b_self   = (const float*)d_in[7];
    const float* bn_gamma = (const float*)d_in[8];
    const float* bn_beta  = (const float*)d_in[9];
    const float* w_ih     = (const float*)d_in[10];
    const float* w_hh     = (const float*)d_in[11];
    const float* b_ih     = (const float*)d_in[12];
    const float* b_hh     = (const float*)d_in[13];
    const float* w_cls    = (const float*)d_in[14];
    const float* b_cls    = (const float*)d_in[15];
    const int* src = eidx;            // edge_index row 0
    const int* dst = eidx + E_EDGES;  // edge_index row 1

    // ---- workspace layout (floats) ----
    float* ws = (float*)d_ws;
    size_t off = 0;
    float* deg    = ws + off; off += 100032;                 // N rounded up
    float* agg_e  = ws + off; off += (size_t)N_NODES * H_DIM;
    float* agg_t  = ws + off; off += (size_t)N_NODES * H_DIM;
    float* OUT    = ws + off; off += (size_t)N_NODES * 256;
    float* hseq   = ws + off; off += (size_t)T_STEPS * B_BATCH * H_DIM;
    float* G      = ws + off; off += (size_t)T_STEPS * B_BATCH * 4 * H_DIM;
    float* gates  = ws + off; off += (size_t)B_BATCH * 4 * H_DIM;
    float* hbuf   = ws + off; off += (size_t)B_BATCH * H_DIM;
    float* cbuf   = ws + off; off += (size_t)B_BATCH * H_DIM;
    float* w_pack = ws + off; off += 256 * F_DIM;
    float* bn_sum   = ws + off; off += 128;
    float* bn_sumsq = ws + off; off += 128;
    float* bn_scale = ws + off; off += 128;
    float* bn_shift = ws + off; off += 128;

    // ---- once-per-call precompute ----
    hipMemsetAsync(deg, 0, (size_t)N_NODES * sizeof(float), stream);
    hipMemsetAsync(agg_e, 0, (size_t)N_NODES * H_DIM * sizeof(float), stream);
    k_degree<<<(E_EDGES + 255) / 256, 256, 0, stream>>>(dst, deg);
    k_pack_w<<<(256 * F_DIM) / 256, 256, 0, stream>>>(w_msg, w_self, w_pack);
    {
        long long n = (long long)E_EDGES * H_DIM;
        k_edge_msg<<<(unsigned)((n + 255) / 256), 256, 0, stream>>>(dst, eattr, w_msg, agg_e);
    }

    // ---- T SAGE steps ----
    for (int t = 0; t < T_STEPS; ++t) {
        hipMemcpyAsync(agg_t, agg_e, (size_t)N_NODES * H_DIM * sizeof(float),
                       hipMemcpyDeviceToDevice, stream);
        hipMemsetAsync(bn_sum, 0, 256 * sizeof(float), stream);  // sum + sumsq
        k_gemm_x<<<(N_NODES / 16) * 4 / 8, 256, 0, stream>>>(x, w_pack, OUT, t);
        {
            long long n = (long long)E_EDGES * H_DIM;
            k_scatter<<<(unsigned)((n + 255) / 256), 256, 0, stream>>>(src, dst, OUT, agg_t);
        }
        k_combine<<<(N_NODES + 63) / 64, 128, 0, stream>>>(OUT, agg_t, deg, b_msg, b_self,
                                                           bn_sum, bn_sumsq);
        k_bnfin<<<1, 128, 0, stream>>>(bn_sum, bn_sumsq, bn_gamma, bn_beta,
                                       bn_scale, bn_shift);
        k_focal<<<(B_BATCH * H_DIM) / 256, 256, 0, stream>>>(OUT, ptr, bn_scale, bn_shift,
                                                             hseq + (size_t)t * B_BATCH * H_DIM);
    }

    // ---- LSTM ----
    k_gemm_ih<<<((T_STEPS * B_BATCH / 16) * 8) / 8, 256, 0, stream>>>(hseq, w_ih, G);
    hipMemsetAsync(hbuf, 0, 2 * (size_t)B_BATCH * H_DIM * sizeof(float), stream); // h & c
    for (int t = 0; t < T_STEPS; ++t) {
        k_gemm_gates<<<((B_BATCH / 16) * 8) / 8, 256, 0, stream>>>(
            hbuf, w_hh, G + (size_t)t * B_BATCH * 4 * H_DIM, b_ih, b_hh, gates);
        k_lstm<<<(B_BATCH * H_DIM) / 256, 256, 0, stream>>>(gates, hbuf, cbuf);
    }

    // ---- classifier ----
    k_cls<<<(B_BATCH * C_CLS + 255) / 256, 256, 0, stream>>>(hbuf, w_cls, b_cls, (float*)d_out);
}